// MultiHeadSelfAttention_75419625717961
// MI455X (gfx1250) — compile-verified
//
#include <hip/hip_runtime.h>

// ---------------------------------------------------------------------------
// MultiHeadSelfAttention with dynamic position bias, fp32, WMMA f32 16x16x4,
// flash-style fused attention, async global->LDS tile staging (gfx1250).
// B=4 L=1024 H=768 NH=12 HS=64 DPB=192.
// d_out = [ out (4,1024,768) | raw_attention (4,12,1024,1024) ] fp32
// ---------------------------------------------------------------------------

#define Bq   4
#define Lq   1024
#define Hq   768
#define NHq  12
#define HSq  64
#define DPBq 192
#define BHq  (Bq * NHq)          // 48
#define OUT_ELEMS (Bq * Lq * Hq) // 3145728

typedef __attribute__((ext_vector_type(2))) float f32x2;
typedef __attribute__((ext_vector_type(8))) float f32x8;

__device__ __forceinline__ f32x8 wmma_f32_16x16x4(f32x2 a, f32x2 b, f32x8 c) {
    // D = A(16x4) * B(4x16) + C(16x16), full fp32
    return __builtin_amdgcn_wmma_f32_16x16x4_f32(
        /*neg_a=*/false, a, /*neg_b=*/false, b,
        /*c_mod=*/(short)0, c, /*reuse_a=*/false, /*reuse_b=*/false);
}

// --- gfx1250 async DMA: global -> LDS, tracked by ASYNCcnt ------------------
// Generic pointers to __shared__ carry the LDS byte offset in their low 32
// bits (LDS aperture lives in addr[63:32]), so truncation yields the DS addr.
__device__ __forceinline__ unsigned lds_addr32(const void* p) {
    return (unsigned)(unsigned long long)p;
}
__device__ __forceinline__ void async_ld_b128(unsigned lds_dst,
                                              unsigned long long gbase,
                                              unsigned byte_off) {
    // GVS mode: mem = SADDR(64) + VADDR(32) ; per-lane LDS destination in VDST
    asm volatile("global_load_async_to_lds_b128 %0, %1, %2"
                 :: "v"(lds_dst), "v"(byte_off), "s"(gbase)
                 : "memory");
}
__device__ __forceinline__ void wait_asynccnt0() {
    asm volatile("s_wait_asynccnt 0x0" ::: "memory");
}

// ---------------------------------------------------------------------------
// Workspace layout (floats)
// ---------------------------------------------------------------------------
#define OFF_TABLE  0u            // 2047*12 -> padded 24576
#define OFF_Q      24576u        // 48*1024*64 = 3145728
#define OFF_K      3170304u
#define OFF_V      6316032u      // total 9461760 floats = 37.8 MB

// ---------------------------------------------------------------------------
// Kernel 1: dynamic position bias MLP -> table[2047][12]
// ---------------------------------------------------------------------------
__global__ __launch_bounds__(DPBq) void dpb_kernel(
    const float* __restrict__ w0, const float* __restrict__ b0,
    const float* __restrict__ w1, const float* __restrict__ b1,
    const float* __restrict__ w2, const float* __restrict__ b2,
    float* __restrict__ table)
{
    __shared__ float h0[DPBq];
    __shared__ float h1[DPBq];
    const int r = blockIdx.x;           // 0..2046
    const int d = threadIdx.x;          // 0..191
    const float pos = (float)(r - (Lq - 1));

    float t = pos * w0[d] + b0[d];
    h0[d] = t / (1.0f + __expf(-t));    // silu
    __syncthreads();

    float acc = b1[d];
    for (int e = 0; e < DPBq; ++e) acc += h0[e] * w1[e * DPBq + d];
    h1[d] = acc / (1.0f + __expf(-acc));
    __syncthreads();

    if (d < NHq) {
        float o = b2[d];
        for (int e = 0; e < DPBq; ++e) o += h1[e] * w2[e * NHq + d];
        table[r * NHq + d] = o;
    }
}

// ---------------------------------------------------------------------------
// Kernel 2: QKV projection GEMM (4096x768)@(768x2304)+bias, async LDS staging
// Block 128 (4 waves), 64x64 output tile, K stepped by 16.
// ---------------------------------------------------------------------------
__global__ __launch_bounds__(128) void qkv_kernel(
    const float* __restrict__ x, const float* __restrict__ W,
    const float* __restrict__ in_bias,
    float* __restrict__ Qb, float* __restrict__ Kb, float* __restrict__ Vb)
{
    __shared__ __align__(16) float As[64][16];   // rows x K-slice
    __shared__ __align__(16) float Bs[16][64];   // K-slice x cols

    const int tid  = threadIdx.x;
    const int wave = tid >> 5;
    const int lane = tid & 31;
    const int half = lane >> 4;
    const int m16  = lane & 15;
    const int rbase = blockIdx.y * 64;
    const int nbase = blockIdx.x * 64;

    const unsigned long long xg = (unsigned long long)x;
    const unsigned long long Wg = (unsigned long long)W;

    f32x8 acc[4] = {};

    for (int kb = 0; kb < Hq; kb += 16) {
        // Stage A: 64x16 floats = 256 float4 (2/thread), B: 16x64 (2/thread)
        for (int i = tid; i < 256; i += 128) {
            const int ar = i >> 2, ac4 = (i & 3) * 4;
            async_ld_b128(lds_addr32(&As[ar][ac4]), xg,
                          (unsigned)(((rbase + ar) * Hq + kb + ac4) * 4));
            const int bk = i >> 4, bc4 = (i & 15) * 4;
            async_ld_b128(lds_addr32(&Bs[bk][bc4]), Wg,
                          (unsigned)(((kb + bk) * (3 * Hq) + nbase + bc4) * 4));
        }
        wait_asynccnt0();
        __syncthreads();

        const int am = wave * 16 + m16;
        for (int k4 = 0; k4 < 4; ++k4) {
            const int kk = k4 * 4 + half * 2;
            f32x2 a; a.x = As[am][kk]; a.y = As[am][kk + 1];
            for (int nt = 0; nt < 4; ++nt) {
                f32x2 b; b.x = Bs[kk][nt * 16 + m16]; b.y = Bs[kk + 1][nt * 16 + m16];
                acc[nt] = wmma_f32_16x16x4(a, b, acc[nt]);
            }
        }
        __syncthreads();
    }

    for (int nt = 0; nt < 4; ++nt) {
        for (int i = 0; i < 8; ++i) {
            const int r = rbase + wave * 16 + i + half * 8;   // 0..4095
            const int c = nbase + nt * 16 + m16;              // 0..2303
            const float val = acc[nt][i] + in_bias[c];
            const int head = c / (3 * HSq);
            const int rem  = c % (3 * HSq);
            const int which = rem >> 6;                       // 0=Q 1=K 2=V
            const int dd    = rem & 63;
            const int bb = r >> 10, l = r & (Lq - 1);
            float* dst = (which == 0) ? Qb : (which == 1) ? Kb : Vb;
            dst[(((size_t)(bb * NHq + head)) * Lq + l) * HSq + dd] = val;
        }
    }
}

// ---------------------------------------------------------------------------
// Kernel 3: fused attention (flash-style, online softmax).
// grid (16 mtile, 48 bh), block 128 (4 waves). Each block: 64 query rows.
// Per j-tile (64 keys): S = Q K^T / 8 (WMMA) -> raw_attention (NT stores),
// add dpb/adj/mask, online softmax in registers, P->LDS->A-frag, O += P V.
// ---------------------------------------------------------------------------
__global__ __launch_bounds__(128) void attn_kernel(
    const float* __restrict__ Qb, const float* __restrict__ Kb,
    const float* __restrict__ Vb, const float* __restrict__ table,
    const float* __restrict__ adj, const unsigned char* __restrict__ mask,
    const float* __restrict__ out_bias,
    float* __restrict__ rawS, float* __restrict__ out)
{
    __shared__ __align__(16) float Ks[64][64];      // 16 KB
    __shared__ __align__(16) float Vs[64][64];      // 16 KB
    __shared__ __align__(16) float Ps[4][16][64];   // 16 KB (per-wave P tile)

    const int bh = blockIdx.y;
    const int b = bh / NHq, h = bh % NHq;
    const int mbase = blockIdx.x * 64;
    const int tid  = threadIdx.x;
    const int wave = tid >> 5, lane = tid & 31;
    const int half = lane >> 4, m16 = lane & 15;

    const float* Qp = Qb + (size_t)bh * Lq * HSq;
    const unsigned long long Kg = (unsigned long long)(Kb + (size_t)bh * Lq * HSq);
    const unsigned long long Vg = (unsigned long long)(Vb + (size_t)bh * Lq * HSq);
    const float* adjp = adj + (size_t)b * Lq * Lq;
    const unsigned char* mp = mask + b * Lq;
    float* Sp = rawS + ((size_t)bh << 20);

    // Q fragments for this wave's 16 rows live in registers (row = m16,
    // halves carry even/odd K-pairs per the 16x4 f32 A layout).
    f32x2 qa[16];
    {
        const int qrow = mbase + wave * 16 + m16;
        for (int k4 = 0; k4 < 16; ++k4) {
            const int kk = k4 * 4 + half * 2;
            qa[k4].x = Qp[(size_t)qrow * HSq + kk];
            qa[k4].y = Qp[(size_t)qrow * HSq + kk + 1];
        }
    }

    f32x8 oacc[4] = {};
    float m_old[8], lsum[8];
    for (int vi = 0; vi < 8; ++vi) { m_old[vi] = -3.402823466e38f; lsum[vi] = 0.0f; }

    for (int jb = 0; jb < Lq; jb += 64) {
        __syncthreads();                     // LDS from previous tile is done
        for (int i = tid; i < 1024; i += 128) {   // 2 tiles x 1024 float4
            const int row = i >> 4, c4 = (i & 15) * 4;
            const unsigned off = (unsigned)(((jb + row) * HSq + c4) * 4);
            async_ld_b128(lds_addr32(&Ks[row][c4]), Kg, off);
            async_ld_b128(lds_addr32(&Vs[row][c4]), Vg, off);
        }
        wait_asynccnt0();
        __syncthreads();                     // all waves' K/V tiles landed

        // ---- S tile: 16(M) x 64(N) per wave -------------------------------
        f32x8 sacc[4] = {};
        for (int k4 = 0; k4 < 16; ++k4) {
            const int kk = k4 * 4 + half * 2;
            const f32x2 a = qa[k4];
            for (int nt = 0; nt < 4; ++nt) {
                f32x2 bb; bb.x = Ks[nt * 16 + m16][kk]; bb.y = Ks[nt * 16 + m16][kk + 1];
                sacc[nt] = wmma_f32_16x16x4(a, bb, sacc[nt]);
            }
        }

        // ---- raw store (streaming) + bias/adj/mask ------------------------
        for (int nt = 0; nt < 4; ++nt) {
            const int j = jb + nt * 16 + m16;
            const float tmask = mp[j] ? 0.0f : 1.0f;
            for (int vi = 0; vi < 8; ++vi) {
                const int i = mbase + wave * 16 + vi + 8 * half;
                const float sr = sacc[nt][vi] * 0.125f;   // /sqrt(HS), TEMP=1
                __builtin_nontemporal_store(sr, &Sp[(size_t)i * Lq + j]);
                float vv = sr + table[(i - j + (Lq - 1)) * NHq + h]
                              + adjp[(size_t)i * Lq + j];
                if (tmask != 0.0f) vv = -3.402823466e38f;
                sacc[nt][vi] = vv;
            }
        }

        // ---- online softmax (per row, 16-lane groups share a row set) -----
        for (int vi = 0; vi < 8; ++vi) {
            float tm = fmaxf(fmaxf(sacc[0][vi], sacc[1][vi]),
                             fmaxf(sacc[2][vi], sacc[3][vi]));
            for (int off = 1; off < 16; off <<= 1)
                tm = fmaxf(tm, __shfl_xor(tm, off, 32));
            const float mnew = fmaxf(m_old[vi], tm);
            const float scale = __expf(m_old[vi] - mnew);
            float rs = 0.0f;
            for (int nt = 0; nt < 4; ++nt) {
                const float p = __expf(sacc[nt][vi] - mnew);
                sacc[nt][vi] = p;
                rs += p;
            }
            for (int off = 1; off < 16; off <<= 1)
                rs += __shfl_xor(rs, off, 32);
            lsum[vi] = lsum[vi] * scale + rs;
            m_old[vi] = mnew;
            for (int nt = 0; nt < 4; ++nt) oacc[nt][vi] *= scale;
        }

        // ---- P: C-layout -> per-wave LDS -> A-layout ----------------------
        for (int nt = 0; nt < 4; ++nt)
            for (int vi = 0; vi < 8; ++vi)
                Ps[wave][vi + 8 * half][nt * 16 + m16] = sacc[nt][vi];
        // DS ops are in-order within a wave; Ps is wave-private, no barrier.

        // ---- O += P @ V ---------------------------------------------------
        for (int k4 = 0; k4 < 16; ++k4) {
            const int kk = k4 * 4 + half * 2;
            f32x2 a; a.x = Ps[wave][m16][kk]; a.y = Ps[wave][m16][kk + 1];
            for (int nt = 0; nt < 4; ++nt) {
                f32x2 bb; bb.x = Vs[kk][nt * 16 + m16]; bb.y = Vs[kk + 1][nt * 16 + m16];
                oacc[nt] = wmma_f32_16x16x4(a, bb, oacc[nt]);
            }
        }
    }

    // ---- epilogue: normalize, add out_bias, write (b, l, h*64+d) ----------
    for (int vi = 0; vi < 8; ++vi) {
        const float inv_l = 1.0f / lsum[vi];
        const int l = mbase + wave * 16 + vi + 8 * half;
        for (int nt = 0; nt < 4; ++nt) {
            const int d = nt * 16 + m16;
            out[((size_t)b * Lq + l) * Hq + h * HSq + d] =
                oacc[nt][vi] * inv_l + out_bias[h * HSq + d];
        }
    }
}

// ---------------------------------------------------------------------------
extern "C" void kernel_launch(void* const* d_in, const int* in_sizes, int n_in,
                              void* d_out, int out_size, void* d_ws, size_t ws_size,
                              hipStream_t stream)
{
    const float* x        = (const float*)d_in[0];
    const float* adj      = (const float*)d_in[1];
    const unsigned char* mask = (const unsigned char*)d_in[2];   // jnp bool
    const float* W        = (const float*)d_in[3];
    const float* in_bias  = (const float*)d_in[4];
    const float* out_bias = (const float*)d_in[5];
    const float* w0 = (const float*)d_in[6];
    const float* b0 = (const float*)d_in[7];
    const float* w1 = (const float*)d_in[8];
    const float* b1 = (const float*)d_in[9];
    const float* w2 = (const float*)d_in[10];
    const float* b2 = (const float*)d_in[11];

    float* out  = (float*)d_out;
    float* rawS = out + OUT_ELEMS;              // raw_attention region

    float* ws    = (float*)d_ws;
    float* table = ws + OFF_TABLE;
    float* Qb    = ws + OFF_Q;
    float* Kb    = ws + OFF_K;
    float* Vb    = ws + OFF_V;

    dpb_kernel<<<2 * Lq - 1, DPBq, 0, stream>>>(w0, b0, w1, b1, w2, b2, table);
    qkv_kernel<<<dim3((3 * Hq) / 64, (Bq * Lq) / 64), 128, 0, stream>>>(
        x, W, in_bias, Qb, Kb, Vb);
    attn_kernel<<<dim3(Lq / 64, BHq), 128, 0, stream>>>(
        Qb, Kb, Vb, table, adj, mask, out_bias, rawS, out);
}